// TrainableGraphMemory_11897059410434
// MI455X (gfx1250) — compile-verified
//
#include <hip/hip_runtime.h>
#include <math.h>

// ---------------------------------------------------------------------------
// TrainableGraphMemory pipeline for MI455X (gfx1250, wave32).
// GEMMs use V_WMMA_F32_16X16X4_F32 (fp32 matches reference semantics).
// h-GEMM stages A-panels in LDS (node_emb streams exactly once: 230MB @ 23.3TB/s).
// scores / readout use register blocking to amortize the streaming operand.
// ---------------------------------------------------------------------------

typedef float v2f __attribute__((ext_vector_type(2)));
typedef float v8f __attribute__((ext_vector_type(8)));

#define PPR_ALPHA 0.15f
#define TOPK 5
#define NEG_SLOPE 0.2f
#define WAVE 32

// ---------------- generic fill ----------------
__global__ void k_fill(float* __restrict__ p, float v, long n) {
  long i = (long)blockIdx.x * blockDim.x + threadIdx.x;
  if (i < n) p[i] = v;
}

// ---------------- scores = query @ node_emb^T  (WMMA, 4 B-tiles per wave) --
// one wave per 16-node tile; emb row panel read once, reused for all bt.
__global__ void k_scores_wmma(const float* __restrict__ q,
                              const float* __restrict__ emb,
                              float* __restrict__ scores,
                              int B, int N, int H) {
  const int lane = threadIdx.x & (WAVE - 1);
  const int wid  = blockIdx.x * (blockDim.x >> 5) + (threadIdx.x >> 5);
  const int nTiles = N >> 4;
  if (wid >= nTiles) return;            // wave-uniform exit
  const int nt = wid;
  const int bTiles = B >> 4;
  const int l = lane & 15, half = lane >> 4;
  const float* erow = emb + (size_t)(nt * 16 + l) * H;
  for (int bt0 = 0; bt0 < bTiles; bt0 += 4) {
    v8f zero = {};
    v8f acc[4] = {zero, zero, zero, zero};
    for (int k0 = 0; k0 < H; k0 += 4) {
      v2f b = *(const v2f*)(erow + k0 + 2 * half);   // K = {2h, 2h+1}
      __builtin_prefetch(erow + k0 + 32, 0, 3);      // global_prefetch_b8
#pragma unroll
      for (int j = 0; j < 4; ++j) {
        int bt = bt0 + j; if (bt >= bTiles) bt = bTiles - 1;  // clamp, EXEC stays full
        v2f a = *(const v2f*)(q + (size_t)(bt * 16 + l) * H + k0 + 2 * half);
        acc[j] = __builtin_amdgcn_wmma_f32_16x16x4_f32(false, a, false, b,
                                                       (short)0, acc[j], false, false);
      }
    }
#pragma unroll
    for (int j = 0; j < 4; ++j) {
      int bt = bt0 + j;
      if (bt < bTiles) {
        float* srow = scores + (size_t)(bt * 16 + 8 * half) * N + nt * 16 + l;
#pragma unroll
        for (int r = 0; r < 8; ++r) srow[(size_t)r * N] = acc[j][r];
      }
    }
  }
}

// ---------------- top-5 per batch row (two-stage LDS merge) ----------------
__global__ void k_topk(const float* __restrict__ scores, int* __restrict__ seeds,
                       int N) {
  const int b = blockIdx.x;
  const int tid = threadIdx.x;
  const int T = blockDim.x;  // 256
  __shared__ float sv[256 * TOPK];
  __shared__ int   si[256 * TOPK];
  __shared__ float sv2[32 * TOPK];
  __shared__ int   si2[32 * TOPK];
  float lv[TOPK];
  int   li[TOPK];
#pragma unroll
  for (int i = 0; i < TOPK; ++i) { lv[i] = -3.4e38f; li[i] = 0; }
  const float* rowp = scores + (size_t)b * N;
  for (int n = tid; n < N; n += T) {
    float v = rowp[n];
    if (v > lv[TOPK - 1]) {
      int pos = TOPK - 1;
      while (pos > 0 && v > lv[pos - 1]) {
        lv[pos] = lv[pos - 1]; li[pos] = li[pos - 1]; --pos;
      }
      lv[pos] = v; li[pos] = n;
    }
  }
#pragma unroll
  for (int i = 0; i < TOPK; ++i) { sv[tid * TOPK + i] = lv[i]; si[tid * TOPK + i] = li[i]; }
  __syncthreads();
  // stage 2: 32 threads each merge 8 lists (40 entries) -> top5
  if (tid < 32) {
#pragma unroll
    for (int i = 0; i < TOPK; ++i) { lv[i] = -3.4e38f; li[i] = 0; }
    for (int j = tid * 8 * TOPK; j < (tid + 1) * 8 * TOPK; ++j) {
      float v = sv[j];
      if (v > lv[TOPK - 1]) {
        int pos = TOPK - 1;
        while (pos > 0 && v > lv[pos - 1]) {
          lv[pos] = lv[pos - 1]; li[pos] = li[pos - 1]; --pos;
        }
        lv[pos] = v; li[pos] = si[j];
      }
    }
#pragma unroll
    for (int i = 0; i < TOPK; ++i) { sv2[tid * TOPK + i] = lv[i]; si2[tid * TOPK + i] = li[i]; }
  }
  __syncthreads();
  if (tid == 0) {
    for (int k = 0; k < TOPK; ++k) {
      float best = -3.4e38f; int bi = 0; int bj = 0;
      for (int j = 0; j < 32 * TOPK; ++j)
        if (sv2[j] > best) { best = sv2[j]; bi = si2[j]; bj = j; }
      seeds[b * TOPK + k] = bi;
      sv2[bj] = -3.4e38f;
    }
  }
}

// ---------------- seed injection (p stored transposed: p_t[N][B]) ----------
__global__ void k_seed(const int* __restrict__ seeds, float* __restrict__ pt,
                       int B) {
  int i = blockIdx.x * blockDim.x + threadIdx.x;
  if (i >= B * TOPK) return;
  int b = i / TOPK;
  int s = seeds[i];
  pt[(size_t)s * B + b] = 1.0f;
}

// ---------------- v_edge = w_edge @ att_edge ----------------
__global__ void k_vedge(const float* __restrict__ we, const float* __restrict__ ae,
                        float* __restrict__ v, int H) {
  int i = blockIdx.x * blockDim.x + threadIdx.x;
  if (i >= H) return;
  float s = 0.f;
  for (int j = 0; j < H; ++j) s += we[(size_t)i * H + j] * ae[j];
  v[i] = s;
}

// ---------------- per-edge: w=||ea||, a_edge=ea.v, deg scatter -------------
__global__ void k_edgew(const float* __restrict__ ea, const float* __restrict__ v,
                        const int* __restrict__ row, float* __restrict__ w,
                        float* __restrict__ aedge, float* __restrict__ deg,
                        int E, int H) {
  const int lane = threadIdx.x & (WAVE - 1);
  const int e = blockIdx.x * (blockDim.x >> 5) + (threadIdx.x >> 5);
  if (e >= E) return;
  const float* r = ea + (size_t)e * H;
  float ss = 0.f, dv = 0.f;
  for (int j = lane; j < H; j += WAVE) {
    float x = r[j];
    ss += x * x;
    dv += x * v[j];
  }
#pragma unroll
  for (int o = 16; o > 0; o >>= 1) {
    ss += __shfl_down(ss, o, WAVE);
    dv += __shfl_down(dv, o, WAVE);
  }
  if (lane == 0) {
    float wv = sqrtf(ss);
    w[e] = wv;
    aedge[e] = dv;
    atomicAdd(&deg[row[e]], wv);
  }
}

__global__ void k_wn(const float* __restrict__ w, const float* __restrict__ deg,
                     const int* __restrict__ row, float* __restrict__ wn, int E) {
  int e = blockIdx.x * blockDim.x + threadIdx.x;
  if (e >= E) return;
  wn[e] = w[e] / (deg[row[e]] + 1e-12f);
}

// ---------------- PPR scatter: p_next[row][b] += wn * p[col][b] ------------
__global__ void k_ppr_scatter(const float* __restrict__ pt, float* __restrict__ pn,
                              const float* __restrict__ wn,
                              const int* __restrict__ row, const int* __restrict__ col,
                              int E, int B) {
  const int lane = threadIdx.x & (WAVE - 1);
  const int nw = (B + WAVE - 1) / WAVE;
  const long slot = (long)blockIdx.x * (blockDim.x >> 5) + (threadIdx.x >> 5);
  if (slot >= (long)E * nw) return;
  const int e = (int)(slot / nw);
  const int bb = (int)(slot - (long)e * nw) * WAVE + lane;
  if (bb >= B) return;
  float val = wn[e] * pt[(size_t)col[e] * B + bb];
  atomicAdd(&pn[(size_t)row[e] * B + bb], val);
}

__global__ void k_ppr_combine(float* __restrict__ pt, const float* __restrict__ pn,
                              long n) {
  long i = (long)blockIdx.x * blockDim.x + threadIdx.x;
  if (i >= n) return;
  pt[i] = PPR_ALPHA * pt[i] + (1.0f - PPR_ALPHA) * pn[i];
}

// ---------------- h = node_emb @ w_gat  (WMMA + LDS A-panel) ---------------
// one block (8 waves) per 16-node tile; A panel (16 x H) staged in LDS once;
// each wave covers ht = wv, wv+8, wv+16. node_emb streams exactly once.
__global__ void k_h_wmma_lds(const float* __restrict__ emb,
                             const float* __restrict__ wg,
                             float* __restrict__ hout, int N, int H) {
  extern __shared__ float sA[];            // [16][H]
  const int tid = threadIdx.x;
  const int lane = tid & (WAVE - 1);
  const int wv = tid >> 5;                 // 0..7
  const int nt = blockIdx.x;
  const int hTiles = H >> 4;
  // stage A panel (contiguous 16*H block of node_emb)
  const float* src = emb + (size_t)nt * 16 * H;
  for (int i = tid; i < 16 * H; i += blockDim.x) {
    __builtin_prefetch(src + i + 2048, 0, 1);
    sA[i] = src[i];
  }
  __syncthreads();
  const int l = lane & 15, half = lane >> 4;
  for (int hti = wv; hti < hTiles; hti += 8) {   // wave-uniform
    v8f acc = {};
    for (int k0 = 0; k0 < H; k0 += 4) {
      v2f a = *(const v2f*)(sA + (size_t)l * H + k0 + 2 * half);  // ds_load_b64
      v2f b;
      b.x = wg[(size_t)(k0 + 2 * half + 0) * H + hti * 16 + l];
      b.y = wg[(size_t)(k0 + 2 * half + 1) * H + hti * 16 + l];
      acc = __builtin_amdgcn_wmma_f32_16x16x4_f32(false, a, false, b,
                                                  (short)0, acc, false, false);
    }
    float* hrow = hout + (size_t)(nt * 16 + 8 * half) * H + hti * 16 + l;
#pragma unroll
    for (int r = 0; r < 8; ++r) hrow[(size_t)r * H] = acc[r];
  }
}

// ---------------- a_src / a_dst (wave per node) ----------------
__global__ void k_srcdst(const float* __restrict__ h, const float* __restrict__ as,
                         const float* __restrict__ ad, float* __restrict__ osrc,
                         float* __restrict__ odst, int N, int H) {
  const int lane = threadIdx.x & (WAVE - 1);
  const int node = blockIdx.x * (blockDim.x >> 5) + (threadIdx.x >> 5);
  if (node >= N) return;
  const float* r = h + (size_t)node * H;
  float s0 = 0.f, s1 = 0.f;
  for (int j = lane; j < H; j += WAVE) {
    float x = r[j];
    s0 += x * as[j];
    s1 += x * ad[j];
  }
#pragma unroll
  for (int o = 16; o > 0; o >>= 1) {
    s0 += __shfl_down(s0, o, WAVE);
    s1 += __shfl_down(s1, o, WAVE);
  }
  if (lane == 0) { osrc[node] = s0; odst[node] = s1; }
}

// ---------------- segment softmax over col ----------------
__device__ __forceinline__ unsigned fkey(float f) {
  unsigned u = __float_as_uint(f);
  return (u & 0x80000000u) ? ~u : (u | 0x80000000u);
}
__device__ __forceinline__ float fkey_inv(unsigned u) {
  unsigned v = (u & 0x80000000u) ? (u & 0x7fffffffu) : ~u;
  return __uint_as_float(v);
}

__global__ void k_logits(const float* __restrict__ asrc, const float* __restrict__ adst,
                         const float* __restrict__ aedge,
                         const int* __restrict__ row, const int* __restrict__ col,
                         float* __restrict__ logits, unsigned* __restrict__ segmax,
                         int E) {
  int e = blockIdx.x * blockDim.x + threadIdx.x;
  if (e >= E) return;
  float lg = asrc[row[e]] + adst[col[e]] + aedge[e];
  lg = (lg >= 0.f) ? lg : NEG_SLOPE * lg;
  logits[e] = lg;
  atomicMax(&segmax[col[e]], fkey(lg));
}

__global__ void k_exp(const float* __restrict__ logits, const unsigned* __restrict__ segmax,
                      const int* __restrict__ col, float* __restrict__ eexp,
                      float* __restrict__ segsum, int E) {
  int e = blockIdx.x * blockDim.x + threadIdx.x;
  if (e >= E) return;
  float m = fkey_inv(segmax[col[e]]);
  float v = __expf(logits[e] - m);
  eexp[e] = v;
  atomicAdd(&segsum[col[e]], v);
}

__global__ void k_alpha(float* __restrict__ eexp, const float* __restrict__ segsum,
                        const int* __restrict__ col, int E) {
  int e = blockIdx.x * blockDim.x + threadIdx.x;
  if (e >= E) return;
  eexp[e] = eexp[e] / (segsum[col[e]] + 1e-16f);
}

// ---------------- node_feats init = bias broadcast ----------------
__global__ void k_nfinit(float* __restrict__ nf, const float* __restrict__ bias,
                         long NH, int H) {
  long i = (long)blockIdx.x * blockDim.x + threadIdx.x;
  if (i >= NH) return;
  int hh = (int)(i - (i / H) * H);
  nf[i] = bias[hh];
}

// ---------------- message scatter: nf[col] += alpha * h[row] ---------------
__global__ void k_msg(const float* __restrict__ h, const float* __restrict__ alpha,
                      const int* __restrict__ row, const int* __restrict__ col,
                      float* __restrict__ nf, int E, int H) {
  const int lane = threadIdx.x & (WAVE - 1);
  const int e = blockIdx.x * (blockDim.x >> 5) + (threadIdx.x >> 5);
  if (e >= E) return;
  float a = alpha[e];
  const float* src = h + (size_t)row[e] * H;
  float* dst = nf + (size_t)col[e] * H;
  for (int j = lane; j < H; j += WAVE) atomicAdd(&dst[j], a * src[j]);
}

// ---------------- out = p @ node_feats  (WMMA, 4bt x 4ht blocking) ---------
// nf streams once; pt streamed once per ht-group (6x total = 230MB).
__global__ void k_readout_wmma(const float* __restrict__ pt,  // [N][B]
                               const float* __restrict__ nf,  // [N][H]
                               float* __restrict__ out,       // [B][H]
                               int N, int B, int H, int ksplit, int chunk) {
  const int lane = threadIdx.x & (WAVE - 1);
  const int wid = blockIdx.x * (blockDim.x >> 5) + (threadIdx.x >> 5);
  const int hTiles = H >> 4;
  const int hGroups = (hTiles + 3) >> 2;
  if (wid >= hGroups * ksplit) return;   // wave-uniform
  const int kc = wid % ksplit;
  const int hg = wid / ksplit;
  const int bTiles = B >> 4;
  const int kbeg = kc * chunk;
  const int kend = (kbeg + chunk < N) ? kbeg + chunk : N;
  const int l = lane & 15, half = lane >> 4;
  v8f zero = {};
  v8f acc[4][4];
#pragma unroll
  for (int i = 0; i < 4; ++i)
#pragma unroll
    for (int j = 0; j < 4; ++j) acc[i][j] = zero;
  for (int k0 = kbeg; k0 < kend; k0 += 4) {
    v2f a[4], b[4];
#pragma unroll
    for (int j = 0; j < 4; ++j) {
      int bt = (j < bTiles) ? j : bTiles - 1;          // clamp, EXEC stays full
      a[j].x = pt[(size_t)(k0 + 2 * half + 0) * B + bt * 16 + l];
      a[j].y = pt[(size_t)(k0 + 2 * half + 1) * B + bt * 16 + l];
      int ht = hg * 4 + j; if (ht >= hTiles) ht = hTiles - 1;
      b[j].x = nf[(size_t)(k0 + 2 * half + 0) * H + ht * 16 + l];
      b[j].y = nf[(size_t)(k0 + 2 * half + 1) * H + ht * 16 + l];
    }
#pragma unroll
    for (int i = 0; i < 4; ++i)
#pragma unroll
      for (int j = 0; j < 4; ++j)
        acc[i][j] = __builtin_amdgcn_wmma_f32_16x16x4_f32(false, a[i], false, b[j],
                                                          (short)0, acc[i][j], false, false);
  }
#pragma unroll
  for (int i = 0; i < 4; ++i) {
    if (i < bTiles) {
#pragma unroll
      for (int j = 0; j < 4; ++j) {
        int ht = hg * 4 + j;
        if (ht < hTiles) {
#pragma unroll
          for (int r = 0; r < 8; ++r)
            atomicAdd(&out[(size_t)(i * 16 + 8 * half + r) * H + ht * 16 + l],
                      acc[i][j][r]);
        }
      }
    }
  }
}

// ===========================================================================
extern "C" void kernel_launch(void* const* d_in, const int* in_sizes, int n_in,
                              void* d_out, int out_size, void* d_ws, size_t ws_size,
                              hipStream_t stream) {
  const float* query     = (const float*)d_in[0];
  const float* node_emb  = (const float*)d_in[1];
  const float* edge_attr = (const float*)d_in[2];
  const float* w_gat     = (const float*)d_in[3];
  const float* att_src   = (const float*)d_in[4];
  const float* att_dst   = (const float*)d_in[5];
  const float* w_edge    = (const float*)d_in[6];
  const float* att_edge  = (const float*)d_in[7];
  const float* bias      = (const float*)d_in[8];
  const int*   row       = (const int*)d_in[9];
  const int*   col       = (const int*)d_in[10];
  float* out = (float*)d_out;

  const int H = in_sizes[4];
  const int B = in_sizes[0] / H;
  const int N = in_sizes[1] / H;
  const int E = in_sizes[9];

  // -------- workspace layout (floats) --------
  float* ws = (float*)d_ws;
  size_t off = 0;
  float* scores = ws + off; off += (size_t)B * N;
  float* h      = ws + off; off += (size_t)N * H;
  float* nf     = ws + off; off += (size_t)N * H;
  float* p_t    = ws + off; off += (size_t)N * B;
  float* p_nx   = ws + off; off += (size_t)N * B;
  float* w_e    = ws + off; off += (size_t)E;
  float* wn_e   = ws + off; off += (size_t)E;
  float* a_edge = ws + off; off += (size_t)E;
  float* logits = ws + off; off += (size_t)E;
  float* eexp   = ws + off; off += (size_t)E;
  float* deg    = ws + off; off += (size_t)N;
  float* a_src  = ws + off; off += (size_t)N;
  float* a_dst  = ws + off; off += (size_t)N;
  unsigned* segmax = (unsigned*)(ws + off); off += (size_t)N;
  float* segsum = ws + off; off += (size_t)N;
  float* v_edge = ws + off; off += (size_t)H;
  int*   seeds  = (int*)(ws + off); off += (size_t)B * TOPK;
  if (off * sizeof(float) > ws_size) return;  // workspace too small

  const int T = 256;                // 8 waves per block
  const int WPB = T / WAVE;
  auto blocksFor  = [](long n, int t) { return (int)((n + t - 1) / t); };
  auto blocksForW = [&](long waves) { return (int)((waves + WPB - 1) / WPB); };

  // 1) entity scores (WMMA, 4 bt tiles per wave; emb streams once)
  k_scores_wmma<<<blocksForW((long)(N >> 4)), T, 0, stream>>>(query, node_emb, scores, B, N, H);

  // 2) top-5 seeds
  k_topk<<<B, 256, 0, stream>>>(scores, seeds, N);

  // 3) zero-init PPR state + accumulators + output
  k_fill<<<blocksFor((long)N * B, T), T, 0, stream>>>(p_t, 0.f, (long)N * B);
  k_fill<<<blocksFor((long)N, T), T, 0, stream>>>(deg, 0.f, (long)N);
  k_fill<<<blocksFor((long)N, T), T, 0, stream>>>((float*)segmax, 0.f, (long)N); // key 0 == -inf
  k_fill<<<blocksFor((long)N, T), T, 0, stream>>>(segsum, 0.f, (long)N);
  k_fill<<<blocksFor((long)out_size, T), T, 0, stream>>>(out, 0.f, (long)out_size);
  k_seed<<<blocksFor((long)B * TOPK, T), T, 0, stream>>>(seeds, p_t, B);

  // 4) edge prep: v_edge = w_edge @ att_edge; fused w/a_edge/deg pass; wn
  k_vedge<<<blocksFor((long)H, T), T, 0, stream>>>(w_edge, att_edge, v_edge, H);
  k_edgew<<<blocksForW((long)E), T, 0, stream>>>(edge_attr, v_edge, row, w_e, a_edge, deg, E, H);
  k_wn<<<blocksFor((long)E, T), T, 0, stream>>>(w_e, deg, row, wn_e, E);

  // 5) PPR iterations (p transposed [N][B] for contiguous per-edge access)
  const int nwB = (B + WAVE - 1) / WAVE;
  for (int it = 0; it < 3; ++it) {
    k_fill<<<blocksFor((long)N * B, T), T, 0, stream>>>(p_nx, 0.f, (long)N * B);
    k_ppr_scatter<<<blocksForW((long)E * nwB), T, 0, stream>>>(p_t, p_nx, wn_e, row, col, E, B);
    k_ppr_combine<<<blocksFor((long)N * B, T), T, 0, stream>>>(p_t, p_nx, (long)N * B);
  }

  // 6) h = node_emb @ w_gat (WMMA + LDS A-panel; node_emb streams once)
  k_h_wmma_lds<<<N >> 4, T, (size_t)16 * H * sizeof(float), stream>>>(node_emb, w_gat, h, N, H);
  k_srcdst<<<blocksForW((long)N), T, 0, stream>>>(h, att_src, att_dst, a_src, a_dst, N, H);

  // 7) GAT attention: logits -> segment softmax over col
  k_logits<<<blocksFor((long)E, T), T, 0, stream>>>(a_src, a_dst, a_edge, row, col, logits, segmax, E);
  k_exp<<<blocksFor((long)E, T), T, 0, stream>>>(logits, segmax, col, eexp, segsum, E);
  k_alpha<<<blocksFor((long)E, T), T, 0, stream>>>(eexp, segsum, col, E);

  // 8) node_feats = bias + scatter(alpha * h[row]) into col
  k_nfinit<<<blocksFor((long)N * H, T), T, 0, stream>>>(nf, bias, (long)N * H, H);
  k_msg<<<blocksForW((long)E), T, 0, stream>>>(h, eexp, row, col, nf, E, H);

  // 9) out = p @ node_feats (WMMA, 4x4 blocking, K-split + atomic accumulate)
  {
    const int chunk = 240;  // multiple of 4
    const int ksplit = (N + chunk - 1) / chunk;
    const int hGroups = ((H >> 4) + 3) >> 2;
    long waves = (long)hGroups * ksplit;
    k_readout_wmma<<<blocksForW(waves), T, 0, stream>>>(p_t, nf, out, N, B, H, ksplit, chunk);
  }
}